// CGMM_74363063763464
// MI455X (gfx1250) — compile-verified
//
#include <hip/hip_runtime.h>
#include <hip/hip_bf16.h>

// Problem constants (match reference)
#define CC 8
#define MM 32
#define GG 16
#define LL 4

typedef float v2f __attribute__((ext_vector_type(2)));
typedef float v8f __attribute__((ext_vector_type(8)));

// ---------------------------------------------------------------------------
// Generic row-softmax over a strided axis.
// base(r) = (r/d1)*s1 + ((r%d1)/d2)*s2 + (r%d2);  elements at base + i*stride
// ---------------------------------------------------------------------------
__global__ void softmax_rows_kernel(const float* __restrict__ src,
                                    float* __restrict__ dst,
                                    int nrows, int len, int stride,
                                    int d1, int s1, int d2, int s2) {
    int r = blockIdx.x * blockDim.x + threadIdx.x;
    if (r >= nrows) return;
    int base = (r / d1) * s1 + ((r % d1) / d2) * s2 + (r % d2);
    float mx = -3.402823466e38f;
    for (int i = 0; i < len; ++i) mx = fmaxf(mx, src[base + i * stride]);
    float sum = 0.f;
    for (int i = 0; i < len; ++i) sum += expf(src[base + i * stride] - mx);
    float inv = 1.0f / sum;
    for (int i = 0; i < len; ++i)
        dst[base + i * stride] = expf(src[base + i * stride] - mx) * inv;
}

// ---------------------------------------------------------------------------
__global__ void zero_int_kernel(int* __restrict__ p, int n) {
    int t = blockIdx.x * blockDim.x + threadIdx.x;
    if (t < n) p[t] = 0;
}

// in-degree histogram (int atomics, once per call)
__global__ void degree_kernel(const int* __restrict__ dst, int* __restrict__ cnt, int E) {
    int t = blockIdx.x * blockDim.x + threadIdx.x;
    if (t < E) atomicAdd(&cnt[dst[t]], 1);
}

__global__ void invcnt_kernel(const int* __restrict__ cnt, float* __restrict__ inv, int N) {
    int t = blockIdx.x * blockDim.x + threadIdx.x;
    if (t < N) inv[t] = 1.0f / fmaxf((float)cnt[t], 1.0f);
}

// ---------------------------------------------------------------------------
// Single-block exclusive scan over cnt[0..N) -> rowstart[0..N], cursor copy.
// N = 50k -> ~49 chunks of 1024; negligible cost, fully deterministic.
// ---------------------------------------------------------------------------
__global__ __launch_bounds__(1024) void scan_kernel(const int* __restrict__ cnt,
                                                    int* __restrict__ rowstart,
                                                    int* __restrict__ cursor, int N) {
    __shared__ int buf[1024];
    __shared__ int carry;
    if (threadIdx.x == 0) carry = 0;
    __syncthreads();
    for (int base = 0; base < N; base += 1024) {
        int i = base + (int)threadIdx.x;
        int v = (i < N) ? cnt[i] : 0;
        buf[threadIdx.x] = v;
        __syncthreads();
        for (int off = 1; off < 1024; off <<= 1) {
            int t = (threadIdx.x >= (unsigned)off) ? buf[threadIdx.x - off] : 0;
            __syncthreads();
            buf[threadIdx.x] += t;
            __syncthreads();
        }
        int excl = buf[threadIdx.x] - v + carry;
        if (i < N) { rowstart[i] = excl; cursor[i] = excl; }
        __syncthreads();
        if (threadIdx.x == 1023) carry += buf[1023];
        __syncthreads();
    }
    if (threadIdx.x == 0) rowstart[N] = carry;
}

// fill CSR adjacency: eidx[rowstart[d] ...] = src ids of in-edges of d
__global__ void csr_fill_kernel(const int* __restrict__ dst, const int* __restrict__ src,
                                int* __restrict__ cursor, int* __restrict__ eidx, int E) {
    int e = blockIdx.x * blockDim.x + threadIdx.x;
    if (e < E) {
        int p = atomicAdd(&cursor[dst[e]], 1);
        eidx[p] = src[e];
    }
}

// ---------------------------------------------------------------------------
// Layer 0: thread per (node, g).  h layout: [N][G][C]  (C contiguous)
// ---------------------------------------------------------------------------
__global__ void layer0_kernel(const int* __restrict__ x,
                              const float* __restrict__ B0s,  // [C,M,G]
                              const float* __restrict__ Pis,  // [C,G]
                              float* __restrict__ h,          // [N,G,C]
                              float* __restrict__ out,        // [N,L,G]
                              int N) {
    int t = blockIdx.x * blockDim.x + threadIdx.x;
    if (t >= N * GG) return;
    int n = t / GG, g = t % GG;
    int xm = x[n];
    float u[CC];
    float Z = 0.f;
#pragma unroll
    for (int c = 0; c < CC; ++c) {
        u[c] = B0s[(c * MM + xm) * GG + g] * Pis[c * GG + g];
        Z += u[c];
    }
    out[(size_t)n * (LL * GG) + g] = logf(Z);
    float iz = 1.0f / Z;
#pragma unroll
    for (int c = 0; c < CC; ++c)
        h[(size_t)n * (GG * CC) + g * CC + c] = u[c] * iz;
}

// ---------------------------------------------------------------------------
// Pull-mode gather-mean: one wave per node; lane owns a contiguous float4 of
// the node's 128-float [G,C] block.  Each in-edge is one fully-coalesced
// 512B load from the L2-resident h; accumulation in registers; aggr written
// exactly once, pre-scaled by 1/deg.  Zero floating-point atomics.
// ---------------------------------------------------------------------------
__global__ void gather_mean_kernel(const float* __restrict__ h,
                                   const int* __restrict__ rowstart,
                                   const int* __restrict__ eidx,
                                   const float* __restrict__ invc,
                                   float* __restrict__ aggr, int N) {
    int t = blockIdx.x * blockDim.x + threadIdx.x;
    int n = t >> 5;
    int lane = t & 31;
    if (n >= N) return;
    int beg = rowstart[n];
    int end = rowstart[n + 1];
    float4 acc = make_float4(0.f, 0.f, 0.f, 0.f);
    for (int j = beg; j < end; ++j) {
        int s = eidx[j];  // uniform across the wave -> broadcast load
        const float4 v = *(const float4*)(h + (size_t)s * 128 + lane * 4);
        acc.x += v.x; acc.y += v.y; acc.z += v.z; acc.w += v.w;
    }
    float iv = invc[n];
    acc.x *= iv; acc.y *= iv; acc.z *= iv; acc.w *= iv;
    *(float4*)(aggr + (size_t)n * 128 + lane * 4) = acc;
}

// ---------------------------------------------------------------------------
// Graph-layer update with V_WMMA_F32_16X16X4_F32.
// Block = 512 threads = 16 waves; wave w handles mixture g = w for a tile of
// 16 nodes.  QA[node,c] = sum_k aggr[node,k]*Q[c,k] done as two K=4 WMMAs:
//   A (16x4 f32): lane L (L%16 = node row), half = L/16 selects k-pair
//   B (4x16 f32): B[k, c] = Q[c, k, g], columns c>=8 zero-padded
// D (16x16 f32, 8 VGPRs) staged through LDS, then Bn*QA, Z, log, h update.
// All pre-WMMA guards are branchless selects so EXEC stays all-ones.
// ---------------------------------------------------------------------------
__global__ __launch_bounds__(512) void layer_update_kernel(
    const int* __restrict__ x,
    const float* __restrict__ Qg,    // this layer's softmaxed Q [C,C,G]
    const float* __restrict__ Bg,    // this layer's softmaxed B [C,M,G]
    const float* __restrict__ aggr,  // [N,G,C] (already mean-scaled)
    float* __restrict__ h,           // [N,G,C]
    float* __restrict__ out,         // [N,L,G]
    int N, int lout) {
    __shared__ float qa[GG * 16 * CC];  // [wave][node-in-tile][c] = 8 KB

    const int tid = threadIdx.x;
    const int w = tid >> 5;    // wave id == g
    const int lane = tid & 31;
    const int half = lane >> 4;
    const int col = lane & 15;  // node row (A) / c column (B,D)
    const int g = w;

    const int node = blockIdx.x * 16 + col;
    const int nn = node < N ? node : N - 1;
    const float sc = node < N ? 1.0f : 0.0f;  // tail mask only
    const size_t abase = (size_t)nn * (GG * CC) + g * CC;
    const int k0 = half * 2;  // k-pair held by this half-wave

    v2f a0, a1, b0, b1;
    a0.x = aggr[abase + k0 + 0] * sc;
    a0.y = aggr[abase + k0 + 1] * sc;
    a1.x = aggr[abase + k0 + 4] * sc;
    a1.y = aggr[abase + k0 + 5] * sc;

    const int qc = col & 7;  // clamped c for safe loads; padded lanes -> 0
    const float cm = (col < CC) ? 1.0f : 0.0f;
    b0.x = Qg[(qc * CC + k0 + 0) * GG + g] * cm;
    b0.y = Qg[(qc * CC + k0 + 1) * GG + g] * cm;
    b1.x = Qg[(qc * CC + k0 + 4) * GG + g] * cm;
    b1.y = Qg[(qc * CC + k0 + 5) * GG + g] * cm;

    v8f acc = {0.f, 0.f, 0.f, 0.f, 0.f, 0.f, 0.f, 0.f};
    acc = __builtin_amdgcn_wmma_f32_16x16x4_f32(false, a0, false, b0,
                                                (short)0, acc, false, false);
    acc = __builtin_amdgcn_wmma_f32_16x16x4_f32(false, a1, false, b1,
                                                (short)0, acc, false, false);

    // D VGPR r: rows m = r (lanes 0-15) / r+8 (lanes 16-31), column = col.
    if (col < CC) {
#pragma unroll
        for (int r = 0; r < 8; ++r)
            qa[(w * 16 + (r + half * 8)) * CC + col] = acc[r];
    }
    __syncthreads();

    if (lane < 16) {
        int n2 = blockIdx.x * 16 + lane;
        if (n2 < N) {
            int xm = x[n2];
            float u[CC];
            float Z = 0.f;
#pragma unroll
            for (int c = 0; c < CC; ++c) {
                float bn = Bg[(c * MM + xm) * GG + g];
                u[c] = bn * qa[(w * 16 + lane) * CC + c];
                Z += u[c];
            }
            out[(size_t)n2 * (LL * GG) + lout * GG + g] = logf(Z);
            float iz = 1.0f / Z;
#pragma unroll
            for (int c = 0; c < CC; ++c)
                h[(size_t)n2 * (GG * CC) + g * CC + c] = u[c] * iz;
        }
    }
}

// ---------------------------------------------------------------------------
extern "C" void kernel_launch(void* const* d_in, const int* in_sizes, int n_in,
                              void* d_out, int out_size, void* d_ws, size_t ws_size,
                              hipStream_t stream) {
    const int* x = (const int*)d_in[0];
    const int* ei = (const int*)d_in[1];
    const float* lB0 = (const float*)d_in[2];
    const float* lPi = (const float*)d_in[3];
    const float* lQ = (const float*)d_in[4];
    const float* lB = (const float*)d_in[5];
    float* out = (float*)d_out;

    const int N = in_sizes[0];
    const int E = in_sizes[1] / 2;
    const int* dst = ei;      // edge_index[0]
    const int* srcv = ei + E; // edge_index[1]

    // ---- workspace layout ----
    float* ws = (float*)d_ws;
    size_t off = 0;
    float* B0s = ws + off; off += CC * MM * GG;              // 4096
    float* Pis = ws + off; off += CC * GG;                   // 128
    float* Qs  = ws + off; off += (LL - 1) * CC * CC * GG;   // 3072
    float* Bs  = ws + off; off += (LL - 1) * CC * MM * GG;   // 12288
    float* invc = ws + off; off += N;
    int* cnt      = (int*)(ws + off); off += N;
    int* rowstart = (int*)(ws + off); off += N + 1;
    int* cursor   = (int*)(ws + off); off += N;
    int* eidx     = (int*)(ws + off); off += E;
    off = (off + 15) & ~(size_t)15;
    float* h    = ws + off; off += (size_t)N * GG * CC;
    float* aggr = ws + off; off += (size_t)N * GG * CC;
    (void)ws_size; (void)n_in; (void)out_size;

    // ---- parameter softmaxes ----
    // B0: [C,M,G] over M: rows (c,g): base = c*M*G + g
    softmax_rows_kernel<<<1, 128, 0, stream>>>(lB0, B0s, CC * GG, MM, GG,
                                               CC * GG, 0, GG, MM * GG);
    // Pi: [C,G] over C: rows g: base = g
    softmax_rows_kernel<<<1, 32, 0, stream>>>(lPi, Pis, GG, CC, GG,
                                              GG, 0, GG, 0);
    // Q: [L-1,C,C,G] over axis-1 (i): rows (l,k,g): base = l*CCG + k*G + g
    softmax_rows_kernel<<<3, 128, 0, stream>>>(lQ, Qs, (LL - 1) * CC * GG, CC,
                                               CC * GG, CC * GG, CC * CC * GG,
                                               1, 1);
    // B: [L-1,C,M,G] over M: rows (l,c,g): base = l*CMG + c*M*G + g
    softmax_rows_kernel<<<3, 128, 0, stream>>>(lB, Bs, (LL - 1) * CC * GG, MM,
                                               GG, CC * GG, CC * MM * GG,
                                               GG, MM * GG);

    // ---- CSR build (once per call) ----
    zero_int_kernel<<<(N + 255) / 256, 256, 0, stream>>>(cnt, N);
    degree_kernel<<<(E + 255) / 256, 256, 0, stream>>>(dst, cnt, E);
    invcnt_kernel<<<(N + 255) / 256, 256, 0, stream>>>(cnt, invc, N);
    scan_kernel<<<1, 1024, 0, stream>>>(cnt, rowstart, cursor, N);
    csr_fill_kernel<<<(E + 255) / 256, 256, 0, stream>>>(dst, srcv, cursor, eidx, E);

    // ---- layer 0 ----
    layer0_kernel<<<(N * GG + 255) / 256, 256, 0, stream>>>(x, B0s, Pis, h, out, N);

    // ---- graph layers ----
    const int gatherBlocks = (int)(((size_t)N * 32 + 255) / 256);
    for (int l = 0; l < LL - 1; ++l) {
        gather_mean_kernel<<<gatherBlocks, 256, 0, stream>>>(h, rowstart, eidx,
                                                             invc, aggr, N);
        layer_update_kernel<<<(N + 15) / 16, 512, 0, stream>>>(
            x, Qs + (size_t)l * CC * CC * GG, Bs + (size_t)l * CC * MM * GG,
            aggr, h, out, N, l + 1);
    }
}